// CondNodeFeat_79517024518204
// MI455X (gfx1250) — compile-verified
//
#include <hip/hip_runtime.h>

typedef float v2f __attribute__((ext_vector_type(2)));
typedef float v8f __attribute__((ext_vector_type(8)));
typedef int   v4i __attribute__((ext_vector_type(4)));

#define BB   8
#define NN   2048
#define KK   32
#define CC   256
#define CONDD 512
#define MTOT (BB * NN)          // 16384 rows
#define LN_EPS 1e-5f
#define WT_STRIDE 34            // padded k-stride for transposed W chunk (even -> v2f aligned)

#if __has_builtin(__builtin_amdgcn_global_load_async_to_lds_b128)
#define HAVE_ASYNC_LDS 1
#else
#define HAVE_ASYNC_LDS 0
#endif

#define GLOBAL_AS __attribute__((address_space(1)))
#define LDS_AS    __attribute__((address_space(3)))

__device__ __forceinline__ void wait_async_zero() {
#if __has_builtin(__builtin_amdgcn_s_wait_asynccnt)
    __builtin_amdgcn_s_wait_asynccnt(0);
#else
    asm volatile("s_wait_asynccnt 0" ::: "memory");
#endif
}

// ---------------------------------------------------------------------------
// Kernel 1: FiLM conditioning projection  gb[b, 0:256]=gamma+1, gb[b,256:512]=beta
// gb = cond @ W_cond + b_cond ;  [8,512] x [512,512]
// ---------------------------------------------------------------------------
__global__ __launch_bounds__(256) void cond_proj_kernel(
    const float* __restrict__ cond,   // [8, 512]
    const float* __restrict__ Wc,     // [512, 512]
    const float* __restrict__ bc,     // [512]
    float* __restrict__ gb)           // [8, 512] out
{
    const int t = blockIdx.x * 256 + threadIdx.x;   // 0..4095
    const int b = t >> 9;                           // / 512
    const int j = t & 511;
    const float* crow = cond + b * CONDD;
    float acc = bc[j];
#pragma unroll 4
    for (int k = 0; k < CONDD; ++k)
        acc = fmaf(crow[k], Wc[k * (2 * CC) + j], acc);
    if (j < CC) acc += 1.0f;     // gamma = gamma + 1
    gb[b * (2 * CC) + j] = acc;
}

// ---------------------------------------------------------------------------
// Kernel 2: h = node @ W_film + b_film ; LayerNorm(C) ; FiLM ; ReLU
// One block = 16 rows x 256 cols. 8 waves, each owns two 16x16 N-tiles.
// fp32 WMMA (V_WMMA_F32_16X16X4_F32). A tile staged via async global->LDS;
// W staged in 32-k-row chunks TRANSPOSED so B fragments are contiguous b64.
// ---------------------------------------------------------------------------
__global__ __launch_bounds__(256) void film_gemm_ln_kernel(
    const float* __restrict__ node,   // [16384, 256]
    const float* __restrict__ Wf,     // [256, 256]
    const float* __restrict__ bf,     // [256]
    const float* __restrict__ gb,     // [8, 512] gamma(+1)|beta
    float* __restrict__ nf)           // [16384, 256] out (new_node_feats)
{
    __shared__ __align__(16) float sA[16 * CC];          // 16 KB  A tile (16 x K=256)
    __shared__ __align__(16) float sBT[CC * WT_STRIDE];  // ~34 KB  W chunk transposed [j][k]; reused as H tile
    __shared__ float sred[2 * 256];                      // partial sums / sumsq
    __shared__ float sstat[2 * 16];                      // mu / rsigma per row

    const int t     = threadIdx.x;
    const int w     = t >> 5;         // wave id 0..7 (wave32)
    const int lane  = t & 31;
    const int lrow  = lane & 15;      // M (A frag) / N (B frag) index within tile
    const int lhalf = lane >> 4;      // selects K pair within the 16x16x4 step
    const int m0    = blockIdx.x * 16;
    const int j0    = w * 32;         // this wave covers cols [j0, j0+32)

    // ---- stage A tile (16 x 256 floats): async DMA global -> LDS ----
    {
        const float4* g = (const float4*)(node + (long)m0 * CC);
        float4* s = (float4*)sA;
#if HAVE_ASYNC_LDS
#pragma unroll
        for (int i = 0; i < 4; ++i)
            __builtin_amdgcn_global_load_async_to_lds_b128(
                (GLOBAL_AS v4i*)(g + t + i * 256),
                (LDS_AS    v4i*)(s + t + i * 256), 0, 0);
        wait_async_zero();
#else
#pragma unroll
        for (int i = 0; i < 4; ++i) s[t + i * 256] = g[t + i * 256];
#endif
    }

    v8f acc0 = {0.f, 0.f, 0.f, 0.f, 0.f, 0.f, 0.f, 0.f};
    v8f acc1 = acc0;

    for (int kc = 0; kc < CC; kc += 32) {
        __syncthreads();              // sBT consumed by previous chunk / sA ready
        {
            // load 32 k-rows x 256 cols of W, store transposed: sBT[j*34 + k]
            const float4* g = (const float4*)(Wf + (long)kc * CC);
#pragma unroll
            for (int i = 0; i < 8; ++i) {
                const int linear = t + i * 256;       // 0..2047 float4 slots
                const int k  = linear >> 6;           // k-row within chunk (0..31)
                const int j  = (linear & 63) * 4;     // starting column
                const float4 v = g[linear];
                sBT[(j + 0) * WT_STRIDE + k] = v.x;
                sBT[(j + 1) * WT_STRIDE + k] = v.y;
                sBT[(j + 2) * WT_STRIDE + k] = v.z;
                sBT[(j + 3) * WT_STRIDE + k] = v.w;
            }
        }
        __syncthreads();

#pragma unroll
        for (int k4 = 0; k4 < 8; ++k4) {
            const int kloc = k4 * 4 + lhalf * 2;     // K offset within chunk for this half-wave
            // A fragment: lane holds A[M=lrow][kc+kloc .. +1]  (ISA 16x4 fp32 layout)
            v2f a  = *(const v2f*)&sA[lrow * CC + kc + kloc];
            // B fragments: contiguous in transposed layout -> single ds_load_b64 each
            v2f b0 = *(const v2f*)&sBT[(j0 + lrow) * WT_STRIDE + kloc];
            v2f b1 = *(const v2f*)&sBT[(j0 + 16 + lrow) * WT_STRIDE + kloc];
            acc0 = __builtin_amdgcn_wmma_f32_16x16x4_f32(false, a, false, b0,
                                                         (short)0, acc0, false, false);
            acc1 = __builtin_amdgcn_wmma_f32_16x16x4_f32(false, a, false, b1,
                                                         (short)0, acc1, false, false);
        }
    }

    // ---- bias ----
    const float bias0 = bf[j0 + lrow];
    const float bias1 = bf[j0 + 16 + lrow];
#pragma unroll
    for (int r = 0; r < 8; ++r) { acc0[r] += bias0; acc1[r] += bias1; }

    __syncthreads();                  // everyone done reading sBT; reuse it for H tile
    // D layout: VGPR r holds M = r + 8*lhalf, N = lrow (+ tile offset)
#pragma unroll
    for (int r = 0; r < 8; ++r) {
        const int row = r + lhalf * 8;
        sBT[row * CC + j0 + lrow]      = acc0[r];
        sBT[row * CC + j0 + 16 + lrow] = acc1[r];
    }
    __syncthreads();

    // ---- LayerNorm stats: 16 threads per row, 16-wide segments ----
    {
        const int row = t >> 4, seg = t & 15;
        float s = 0.f, s2 = 0.f;
#pragma unroll
        for (int c = 0; c < 16; ++c) {
            const float v = sBT[row * CC + seg * 16 + c];
            s += v; s2 += v * v;
        }
        sred[t] = s; sred[256 + t] = s2;
    }
    __syncthreads();
    if (t < 16) {
        float s = 0.f, s2 = 0.f;
#pragma unroll
        for (int i = 0; i < 16; ++i) { s += sred[t * 16 + i]; s2 += sred[256 + t * 16 + i]; }
        const float mu  = s * (1.f / 256.f);
        const float var = s2 * (1.f / 256.f) - mu * mu;
        sstat[t]      = mu;
        sstat[16 + t] = rsqrtf(var + LN_EPS);
    }
    __syncthreads();

    // ---- FiLM + ReLU + store (float4, coalesced) ----
    const int bidx = m0 >> 11;        // m0 / 2048 (tile never crosses batch: 2048 % 16 == 0)
    const float4* gmm = (const float4*)(gb + bidx * (2 * CC));
    const float4* bet = (const float4*)(gb + bidx * (2 * CC) + CC);
    const float4* sH4 = (const float4*)sBT;
    float4* outp = (float4*)(nf + (long)m0 * CC);
#pragma unroll
    for (int i = 0; i < 4; ++i) {
        const int linear = t + i * 256;        // 0..1023 float4 slots
        const int row = linear >> 6, c4 = linear & 63;
        const float mu = sstat[row], rs = sstat[16 + row];
        const float4 h  = sH4[linear];
        const float4 g4 = gmm[c4];
        const float4 b4 = bet[c4];
        float4 o;
        o.x = fmaxf(0.f, fmaf((h.x - mu) * rs, g4.x, b4.x));
        o.y = fmaxf(0.f, fmaf((h.y - mu) * rs, g4.y, b4.y));
        o.z = fmaxf(0.f, fmaf((h.z - mu) * rs, g4.z, b4.z));
        o.w = fmaxf(0.f, fmaf((h.w - mu) * rs, g4.w, b4.w));
        outp[linear] = o;
    }
}

// ---------------------------------------------------------------------------
// Kernel 3: nb = sum_k (w*p)[b,n,k] * new[coords[b,n,k], :] ; out = relu(new + nb)
// 64 lanes per row, float4 per lane; gather source is L2-resident (16 MB << 192 MB).
// ---------------------------------------------------------------------------
__global__ __launch_bounds__(256) void gather_agg_kernel(
    const float* __restrict__ nf,       // [16384, 256]
    const float* __restrict__ wts,      // [16384*32]
    const float* __restrict__ prm,      // [16384*32]
    const int*   __restrict__ coords,   // [16384*32], values in [0, 16384)
    float* __restrict__ out)            // [16384, 256]
{
    const int t = threadIdx.x;
    const int r = t >> 6;                       // row within block (0..3)
    const int q = t & 63;                       // float4 slot within row
    const int m = blockIdx.x * 4 + r;
    const long base = (long)m * KK;
    const float4* nf4 = (const float4*)nf;

    float ax = 0.f, ay = 0.f, az = 0.f, aw = 0.f;
    int idx = coords[base];
#pragma unroll 1
    for (int k = 0; k < KK; ++k) {
        const int nidx = (k + 1 < KK) ? coords[base + k + 1] : idx;
        __builtin_prefetch(&nf4[(long)nidx * 64 + q], 0, 0);   // global_prefetch_b8
        const float ew = wts[base + k] * prm[base + k];
        const float4 v = nf4[(long)idx * 64 + q];
        ax = fmaf(ew, v.x, ax);
        ay = fmaf(ew, v.y, ay);
        az = fmaf(ew, v.z, az);
        aw = fmaf(ew, v.w, aw);
        idx = nidx;
    }
    const float4 c = nf4[(long)m * 64 + q];
    float4 o;
    o.x = fmaxf(0.f, c.x + ax);
    o.y = fmaxf(0.f, c.y + ay);
    o.z = fmaxf(0.f, c.z + az);
    o.w = fmaxf(0.f, c.w + aw);
    ((float4*)out)[(long)m * 64 + q] = o;
}

// ---------------------------------------------------------------------------
extern "C" void kernel_launch(void* const* d_in, const int* in_sizes, int n_in,
                              void* d_out, int out_size, void* d_ws, size_t ws_size,
                              hipStream_t stream)
{
    const float* node = (const float*)d_in[0];   // [8,2048,256]
    const float* cond = (const float*)d_in[1];   // [8,512]
    const float* Wc   = (const float*)d_in[2];   // [512,512]
    const float* bc   = (const float*)d_in[3];   // [512]
    const float* Wf   = (const float*)d_in[4];   // [256,256]
    const float* bfm  = (const float*)d_in[5];   // [256]
    const float* wts  = (const float*)d_in[6];   // [8,65536,1]
    const float* prm  = (const float*)d_in[7];   // [8,65536,1]
    const int*   crd  = (const int*)d_in[8];     // [524288]
    float* outp = (float*)d_out;

    char* ws = (char*)d_ws;
    float* gbuf = (float*)ws;                     // 8*512*4   = 16 KB
    float* nfb  = (float*)(ws + 16384);           // 16384*256*4 = 16 MB

    cond_proj_kernel   <<<16,        256, 0, stream>>>(cond, Wc, bc, gbuf);
    film_gemm_ln_kernel<<<MTOT / 16, 256, 0, stream>>>(node, Wf, bfm, gbuf, nfb);
    gather_agg_kernel  <<<MTOT / 4,  256, 0, stream>>>(nfb, wts, prm, crd, outp);
}